// PPO_32813550141590
// MI455X (gfx1250) — compile-verified
//
#include <hip/hip_runtime.h>
#include <hip/hip_bf16.h>
#include <math.h>

// ---------------------------------------------------------------------------
// PPO GAE on MI455X (gfx1250).
// Blocked reverse linear recurrence: adv[s] = a[s]*adv[s+1] + b[s]
//   a[s] = gamma*lambda*nd[s],  b[s] = r[s] - v[s] + gamma*nd[s]*nv[s]
// Layout: (s=t*8+b, e) lives at t*E*B + e*B + b -> per column e the 8 steps of
// one t are 32 contiguous bytes -> two fully-coalesced 16B loads per lane.
// Cache policy: pass1 warms inputs into the 192MB L2 (regular loads); pass3
// consumes them last-use (NT loads), streams `returns` with NT stores, keeps
// raw `adv` regular so the normalize pass hits L2.
// ---------------------------------------------------------------------------

#define T_DIM   512
#define E_DIM   4096
#define B_DIM   8
#define S_DIM   (T_DIM * B_DIM)            // 4096
#define EB      (E_DIM * B_DIM)            // 32768
#define N_TOT   ((size_t)S_DIM * E_DIM)    // 16,777,216
#define TC      16                         // t-steps per chunk
#define C_CHUNKS (T_DIM / TC)              // 32 chunks of L=128 seq steps
#define GAMMA   0.99f
#define LAMBDA  0.95f
#define GLPROD  (GAMMA * LAMBDA)
#define NBLK3   ((E_DIM * C_CHUNKS) / 256) // 512 blocks in compose/apply

typedef __attribute__((ext_vector_type(2))) float    v2f;
typedef __attribute__((ext_vector_type(4))) float    v4f;
typedef __attribute__((ext_vector_type(8))) float    v8f;
typedef __attribute__((ext_vector_type(2))) unsigned v2u;

template <bool NT>
__device__ __forceinline__ void load_pack8(const float* __restrict__ p, float* d) {
  const v4f* q = reinterpret_cast<const v4f*>(p);
  v4f a, b;
  if (NT) { a = __builtin_nontemporal_load(q); b = __builtin_nontemporal_load(q + 1); }
  else    { a = q[0];                          b = q[1]; }
  d[0] = a[0]; d[1] = a[1]; d[2] = a[2]; d[3] = a[3];
  d[4] = b[0]; d[5] = b[1]; d[6] = b[2]; d[7] = b[3];
}

template <bool NT>
__device__ __forceinline__ void load_nd8(const unsigned char* __restrict__ p, float* nd) {
  const v2u* q = reinterpret_cast<const v2u*>(p);    // 8 bool bytes, 8B aligned
  v2u bits = NT ? __builtin_nontemporal_load(q) : q[0];
#pragma unroll
  for (int b = 0; b < 4; ++b) {
    nd[b]     = ((bits[0] >> (8 * b)) & 1u) ? 0.0f : 1.0f;
    nd[4 + b] = ((bits[1] >> (8 * b)) & 1u) ? 0.0f : 1.0f;
  }
}

// --- Pass 1: per-(column, chunk) compose f_s chains into (A, B) -------------
__global__ void __launch_bounds__(256) gae_compose(
    const float* __restrict__ rewards, const float* __restrict__ values,
    const float* __restrict__ last_values, const unsigned char* __restrict__ dones,
    float* __restrict__ wsA, float* __restrict__ wsB)
{
  int gid = blockIdx.x * blockDim.x + threadIdx.x;
  int e = gid & (E_DIM - 1);
  int c = gid >> 12;                       // E_DIM == 4096
  int t_lo = c * TC, t_hi = t_lo + TC - 1;
  size_t col = (size_t)e * B_DIM;

  float vtop = (t_hi == T_DIM - 1) ? last_values[e]
                                   : values[(size_t)(t_hi + 1) * EB + col];
  float A = 1.0f, Bc = 0.0f;
  for (int t = t_hi; t >= t_lo; --t) {
    size_t base = (size_t)t * EB + col;
    // Speculative prefetch of the next pack; non-faulting per ISA, so no guard.
    __builtin_prefetch(rewards + base - EB);
    __builtin_prefetch(values + base - EB);
    __builtin_prefetch(dones + base - EB);
    float rv[8], nd[8], vvx[9];
    load_pack8<false>(rewards + base, rv);
    load_pack8<false>(values + base, vvx);
    load_nd8<false>(dones + base, nd);
    vvx[8] = vtop;
#pragma unroll
    for (int b = 7; b >= 0; --b) {
      float a  = GLPROD * nd[b];
      float bc = rv[b] - vvx[b] + GAMMA * nd[b] * vvx[b + 1];
      Bc = fmaf(a, Bc, bc);                // g <- f_s o g  (s descending)
      A  = a * A;
    }
    vtop = vvx[0];
  }
  wsA[(size_t)c * E_DIM + e] = A;
  wsB[(size_t)c * E_DIM + e] = Bc;
}

// --- Pass 2: per-column reverse prefix over the 32 chunks -------------------
__global__ void __launch_bounds__(256) gae_prefix(
    const float* __restrict__ wsA, const float* __restrict__ wsB,
    float* __restrict__ wsY)
{
  int e = blockIdx.x * blockDim.x + threadIdx.x;
  float y = 0.0f;                          // incoming adv for top chunk
  for (int c = C_CHUNKS - 1; c >= 0; --c) {
    size_t idx = (size_t)c * E_DIM + e;
    wsY[idx] = y;
    y = fmaf(wsA[idx], y, wsB[idx]);       // adv at bottom of chunk c
  }
}

// --- Pass 3: re-run chunks with correct carry; write returns(NT) + raw adv;
//     accumulate (sum, sumsq) -> WMMA wave reduce -> LDS -> block partial ----
__global__ void __launch_bounds__(256) gae_apply(
    const float* __restrict__ rewards, const float* __restrict__ values,
    const float* __restrict__ last_values, const unsigned char* __restrict__ dones,
    const float* __restrict__ wsY,
    float* __restrict__ out_returns, float* __restrict__ out_adv,
    float* __restrict__ wsP)
{
  int gid = blockIdx.x * blockDim.x + threadIdx.x;
  int e = gid & (E_DIM - 1);
  int c = gid >> 12;
  int t_lo = c * TC, t_hi = t_lo + TC - 1;
  size_t col = (size_t)e * B_DIM;

  float vtop = (t_hi == T_DIM - 1) ? last_values[e]
                                   : values[(size_t)(t_hi + 1) * EB + col];
  float adv = wsY[(size_t)c * E_DIM + e];
  float s1 = 0.0f, s2 = 0.0f;

  for (int t = t_hi; t >= t_lo; --t) {
    size_t base = (size_t)t * EB + col;
    __builtin_prefetch(rewards + base - EB);
    __builtin_prefetch(values + base - EB);
    __builtin_prefetch(dones + base - EB);
    float rv[8], nd[8], vvx[9];
    load_pack8<true>(rewards + base, rv);    // last use -> NT
    load_pack8<true>(values + base, vvx);
    load_nd8<true>(dones + base, nd);
    vvx[8] = vtop;
#pragma unroll
    for (int b = 7; b >= 0; --b) {
      float a  = GLPROD * nd[b];
      float bc = rv[b] - vvx[b] + GAMMA * nd[b] * vvx[b + 1];
      adv = fmaf(a, adv, bc);
      size_t oidx = (size_t)(t * B_DIM + b) * E_DIM + e;   // coalesced over e
      __builtin_nontemporal_store(adv + vvx[b], out_returns + oidx); // stream
      out_adv[oidx] = adv;                                 // keep in L2
      s1 += adv;
      s2 = fmaf(adv, adv, s2);
    }
    vtop = vvx[0];
  }

  // Wave-level reduction with V_WMMA_F32_16X16X4_F32 (B = ones):
  // A[m][k] holds lane partials (lane L -> m=L%16, k=2*(L/16)+vgpr);
  // D[m][n] = sum_k A[m][k] -> in-lane component sum + shfl_xor(16) = wave sum.
  v2f aS; aS[0] = s1; aS[1] = 0.0f;
  v2f aQ; aQ[0] = s2; aQ[1] = 0.0f;
  v2f on; on[0] = 1.0f; on[1] = 1.0f;
  v8f dS = {}; v8f dQ = {};
  dS = __builtin_amdgcn_wmma_f32_16x16x4_f32(false, aS, false, on, (short)0, dS, false, false);
  dQ = __builtin_amdgcn_wmma_f32_16x16x4_f32(false, aQ, false, on, (short)0, dQ, false, false);
  float wS = dS[0] + dS[1] + dS[2] + dS[3] + dS[4] + dS[5] + dS[6] + dS[7];
  float wQ = dQ[0] + dQ[1] + dQ[2] + dQ[3] + dQ[4] + dQ[5] + dQ[6] + dQ[7];
  wS += __shfl_xor(wS, 16, 32);
  wQ += __shfl_xor(wQ, 16, 32);

  __shared__ float lS[8], lQ[8];
  int lane = threadIdx.x & 31, wv = threadIdx.x >> 5;
  if (lane == 0) { lS[wv] = wS; lQ[wv] = wQ; }
  __syncthreads();
  if (threadIdx.x == 0) {
    float bS = 0.0f, bQ = 0.0f;
#pragma unroll
    for (int w = 0; w < 8; ++w) { bS += lS[w]; bQ += lQ[w]; }
    wsP[blockIdx.x]         = bS;          // deterministic block partials
    wsP[NBLK3 + blockIdx.x] = bQ;
  }
}

// --- Pass 4: deterministic reduce of 512 block partials to (sum, sumsq) ----
__global__ void __launch_bounds__(256) gae_reduce(
    const float* __restrict__ wsP, double* __restrict__ acc)
{
  __shared__ double lS[256], lQ[256];
  int t = threadIdx.x;
  lS[t] = (double)wsP[t] + (double)wsP[t + 256];
  lQ[t] = (double)wsP[NBLK3 + t] + (double)wsP[NBLK3 + t + 256];
  __syncthreads();
  for (int off = 128; off > 0; off >>= 1) {
    if (t < off) { lS[t] += lS[t + off]; lQ[t] += lQ[t + off]; }
    __syncthreads();
  }
  if (t == 0) { acc[0] = lS[0]; acc[1] = lQ[0]; }
}

// --- Pass 5: normalize advantages in place (NT streaming both ways) ---------
__global__ void __launch_bounds__(256) gae_normalize(
    float* __restrict__ out_adv, const double* __restrict__ acc)
{
  double invN = 1.0 / (double)N_TOT;
  double mean = acc[0] * invN;
  double var  = acc[1] * invN - mean * mean;
  if (var < 0.0) var = 0.0;
  float fmean = (float)mean;
  float scale = 1.0f / ((float)sqrt(var) + 1e-8f);

  size_t i = ((size_t)blockIdx.x * blockDim.x + threadIdx.x) * 4;
  v4f* p = reinterpret_cast<v4f*>(out_adv + i);
  v4f x = __builtin_nontemporal_load(p);   // last use of raw adv
  x[0] = (x[0] - fmean) * scale;
  x[1] = (x[1] - fmean) * scale;
  x[2] = (x[2] - fmean) * scale;
  x[3] = (x[3] - fmean) * scale;
  __builtin_nontemporal_store(x, p);       // final output, stream to HBM
}

extern "C" void kernel_launch(void* const* d_in, const int* in_sizes, int n_in,
                              void* d_out, int out_size, void* d_ws, size_t ws_size,
                              hipStream_t stream) {
  const float* rewards          = (const float*)d_in[0];
  const float* values           = (const float*)d_in[1];
  const float* last_values      = (const float*)d_in[2];   // [E,1] -> flat [E]
  const unsigned char* dones    = (const unsigned char*)d_in[3]; // jax bool = 1 byte

  const size_t CE = (size_t)C_CHUNKS * E_DIM;              // 131072
  char* ws   = (char*)d_ws;
  float* wsA = (float*)ws;
  float* wsB = wsA + CE;
  float* wsY = wsB + CE;
  float* wsP = wsY + CE;                                   // 2*NBLK3 floats
  double* acc = (double*)(wsP + 2 * NBLK3);                // 8B aligned

  float* out_returns = (float*)d_out;
  float* out_adv     = out_returns + N_TOT;

  dim3 blk(256);
  gae_compose<<<NBLK3, blk, 0, stream>>>(rewards, values, last_values, dones, wsA, wsB);
  gae_prefix<<<E_DIM / 256, blk, 0, stream>>>(wsA, wsB, wsY);
  gae_apply<<<NBLK3, blk, 0, stream>>>(rewards, values, last_values, dones, wsY,
                                       out_returns, out_adv, wsP);
  gae_reduce<<<1, blk, 0, stream>>>(wsP, acc);
  gae_normalize<<<(unsigned)(N_TOT / 4 / 256), blk, 0, stream>>>(out_adv, acc);
}